// GATedge_21414706937937
// MI455X (gfx1250) — compile-verified
//
#include <hip/hip_runtime.h>
#include <math.h>

// ---------------------------------------------------------------------------
// GATedge forward for MI455X (gfx1250, wave32).
// Memory-bound (~40MB traffic, ~0.2 GFLOP; HBM 23.3 TB/s -> ~2us; whole
// working set fits in 192MB L2). The einsum
//   b[b,m,f] = sum_o alpha[b,o,m] * feat_src[b,o,f]
// is mapped onto V_WMMA_F32_16X16X4_F32 (16m x 4o alpha A-tile, 4o x 16f
// feat_src B-tile, two N-tiles for F=32). Inner loops are branch-free
// (clamped loads + selects) so loads clause and EXEC stays all-ones.
// ---------------------------------------------------------------------------

typedef float v2f __attribute__((ext_vector_type(2)));
typedef float v8f __attribute__((ext_vector_type(8)));

namespace cfg {
constexpr int B  = 32;    // batch
constexpr int O  = 1000;  // operations (softmax axis, + 1 self row)
constexpr int M  = 100;   // machines
constexpr int F  = 32;    // features
constexpr int Ds = 6;     // raw_opes feature dim
constexpr int Dd = 3;     // raw_mas feature dim
constexpr float NEGV  = -1000000000.0f;
constexpr float SLOPE = 0.2f;

// workspace layout (float offsets)
constexpr size_t off_feat_src = 0;                                   // B*O*F
constexpr size_t off_el       = off_feat_src + (size_t)B * O * F;    // B*O
constexpr size_t off_feat_dst = off_el       + (size_t)B * O;        // B*M*F
constexpr size_t off_er       = off_feat_dst + (size_t)B * M * F;    // B*M
constexpr size_t off_mx       = off_er       + (size_t)B * M;        // B*M
constexpr size_t off_den      = off_mx       + (size_t)B * M;        // B*M
constexpr size_t off_S1       = off_den      + (size_t)B * M;        // B*M
constexpr size_t off_S2       = off_S1       + (size_t)B * M;        // B*M*F

constexpr int MT  = 7;    // ceil(M/16) m-tiles
constexpr int OCN = 8;    // o-chunks
constexpr int OC  = 128;  // o-chunk size (8*128 >= 1000)
} // namespace cfg

__device__ __forceinline__ float wave_sum(float v) {
#pragma unroll
  for (int off = 16; off > 0; off >>= 1) v += __shfl_xor(v, off, 32);
  return v;
}
__device__ __forceinline__ float wave_max(float v) {
#pragma unroll
  for (int off = 16; off > 0; off >>= 1) v = fmaxf(v, __shfl_xor(v, off, 32));
  return v;
}
__device__ __forceinline__ float leaky(float x) {
  return x >= 0.0f ? x : cfg::SLOPE * x;
}
__device__ __forceinline__ void atomic_addf(float* p, float v) {
  __hip_atomic_fetch_add(p, v, __ATOMIC_RELAXED, __HIP_MEMORY_SCOPE_AGENT);
}

// ---------------------------------------------------------------------------
// Kernel 1: tiny GEMM rows: feat[row, f] = sum_d X[row,d]*W[d,f];
//           escore[row]    = sum_f feat[row,f]*attn[f]
// One wave (32 lanes == 32 features) per row.
// ---------------------------------------------------------------------------
__global__ void __launch_bounds__(256) k_feat(const float* __restrict__ X,
                                              const float* __restrict__ W,
                                              const float* __restrict__ attn,
                                              float* __restrict__ feat,
                                              float* __restrict__ escore,
                                              int nrows, int D) {
  int wid  = (blockIdx.x * blockDim.x + threadIdx.x) >> 5;
  int lane = threadIdx.x & 31;
  if (wid >= nrows) return;
  const float* row = X + (size_t)wid * D;
  float acc = 0.0f;
  for (int d = 0; d < D; ++d) acc += row[d] * W[d * cfg::F + lane];
  feat[(size_t)wid * cfg::F + lane] = acc;
  float e = wave_sum(acc * attn[lane]);
  if (lane == 0) escore[wid] = e;
}

// ---------------------------------------------------------------------------
// Kernel 2: per-(b,m) masked softmax stats over the o axis (+ self row ekk).
// One wave per (b,m); lanes stride o; branch-free clamped loads.
// ---------------------------------------------------------------------------
__global__ void __launch_bounds__(256) k_stats(const float* __restrict__ pt,
                                               const int*   __restrict__ adj_all,
                                               const int*   __restrict__ batch_idxes,
                                               const float* __restrict__ el,
                                               const float* __restrict__ er,
                                               const float* __restrict__ W_edge,
                                               const float* __restrict__ attn_l,
                                               float* __restrict__ mx_out,
                                               float* __restrict__ den_out) {
  using namespace cfg;
  int wid  = (blockIdx.x * blockDim.x + threadIdx.x) >> 5;
  int lane = threadIdx.x & 31;
  if (wid >= B * M) return;
  int b = wid / M, m = wid % M;

  float cee  = wave_sum(W_edge[lane] * attn_l[lane]);   // dot(W_edge, attn_l)
  int   bi   = batch_idxes[b];
  float er_m = er[wid];

  float ev[32];
  float lmax = NEGV;
#pragma unroll
  for (int it = 0; it < 32; ++it) {
    int o  = it * 32 + lane;
    int oc = min(o, O - 1);
    int   ad  = adj_all[((size_t)bi * O + oc) * M + m];
    float ptv = pt[((size_t)b * O + oc) * M + m];
    float elv = el[b * O + oc];
    float e   = leaky(elv + er_m + ptv * cee);
    e = (o < O && ad == 1) ? e : NEGV;
    ev[it] = e;
    lmax = fmaxf(lmax, e);
  }
  float ekk   = leaky(2.0f * er_m);
  float mxall = fmaxf(wave_max(lmax), ekk);
  float s = 0.0f;
#pragma unroll
  for (int it = 0; it < 32; ++it) s += __expf(ev[it] - mxall); // exp(NEG)->0
  s = wave_sum(s) + __expf(ekk - mxall);
  if (lane == 0) { mx_out[wid] = mxall; den_out[wid] = s; }
}

// ---------------------------------------------------------------------------
// Kernel 3: WMMA accumulation.
//   S2[b,m,f] += sum_o alpha[b,o,m] * feat_src[b,o,f]   (v_wmma_f32_16x16x4_f32)
//   S1[b,m]   += sum_o alpha[b,o,m] * pt[b,o,m]
// Block = one wave. Grid = B * MT * OCN. A-tile: 16 m x 4 o alphas.
// Branch-free: out-of-range K columns get alpha==0 so B needs no zeroing.
// ---------------------------------------------------------------------------
__global__ void __launch_bounds__(32) k_accum(const float* __restrict__ pt,
                                              const int*   __restrict__ adj_all,
                                              const int*   __restrict__ batch_idxes,
                                              const float* __restrict__ el,
                                              const float* __restrict__ er,
                                              const float* __restrict__ mx,
                                              const float* __restrict__ den,
                                              const float* __restrict__ feat_src,
                                              const float* __restrict__ W_edge,
                                              const float* __restrict__ attn_l,
                                              float* __restrict__ S1,
                                              float* __restrict__ S2) {
  using namespace cfg;
  int blk = blockIdx.x;
  int b   = blk / (MT * OCN);
  int rem = blk % (MT * OCN);
  int mt  = rem / OCN;
  int oc  = rem % OCN;

  int l      = threadIdx.x;
  int mlane  = l & 15;
  int khalf  = l >> 4;         // lanes 0-15 hold K={0,1}; lanes 16-31 K={2,3}
  int m_base = mt * 16;
  int m_glob = m_base + mlane;
  bool mvalid = m_glob < M;
  int m_c    = min(m_glob, M - 1);   // clamped for unconditional loads

  float cee = wave_sum(W_edge[l] * attn_l[l]);
  int   bi  = batch_idxes[b];

  float er_m   = er[b * M + m_c];
  float mx_m   = mx[b * M + m_c];
  float invden = 1.0f / den[b * M + m_c];

  int o_begin = oc * OC;
  int o_end   = min(o_begin + OC, O);

  v8f c0 = {}; v8f c1 = {};
  float s1 = 0.0f;

  for (int k0 = o_begin; k0 < o_end; k0 += 4) {  // uniform loop: EXEC all-1
    int o0  = k0 + 2 * khalf;
    int oc0 = min(o0,     O - 1);
    int oc1 = min(o0 + 1, O - 1);
    size_t r0 = (size_t)b * O + oc0;
    size_t r1 = (size_t)b * O + oc1;

    // unconditional clamped loads -> clause-friendly
    int   ad0 = adj_all[((size_t)bi * O + oc0) * M + m_c];
    int   ad1 = adj_all[((size_t)bi * O + oc1) * M + m_c];
    float pt0 = pt[r0 * M + m_c];
    float pt1 = pt[r1 * M + m_c];
    float el0 = el[b * O + oc0];
    float el1 = el[b * O + oc1];
    const float* f0 = feat_src + r0 * F;
    const float* f1 = feat_src + r1 * F;
    v2f a, b0, b1;
    b0.x = f0[mlane];      b0.y = f1[mlane];
    b1.x = f0[mlane + 16]; b1.y = f1[mlane + 16];

    // prefetch next step's adj/pt rows (global_prefetch_b8)
    int opf = min(o0 + 4, O - 1);
    __builtin_prefetch(&adj_all[((size_t)bi * O + opf) * M + m_c], 0, 1);
    __builtin_prefetch(&pt[((size_t)b * O + opf) * M + m_c], 0, 1);

    bool v0 = mvalid && (o0     < O) && (ad0 == 1);
    bool v1 = mvalid && (o0 + 1 < O) && (ad1 == 1);
    float e0 = leaky(el0 + er_m + pt0 * cee);
    float e1 = leaky(el1 + er_m + pt1 * cee);
    float a0 = v0 ? __expf(e0 - mx_m) * invden : 0.0f;
    float a1 = v1 ? __expf(e1 - mx_m) * invden : 0.0f;
    s1 += a0 * pt0 + a1 * pt1;
    a.x = a0; a.y = a1;

    // D = A(16x4) * B(4x16) + C   (f32)
    c0 = __builtin_amdgcn_wmma_f32_16x16x4_f32(false, a, false, b0, (short)0, c0, false, false);
    c1 = __builtin_amdgcn_wmma_f32_16x16x4_f32(false, a, false, b1, (short)0, c1, false, false);
  }

  // S1: lanes l and l+16 hold partials for the same m
  float s1c = s1 + __shfl_xor(s1, 16, 32);
  if (l < 16 && mvalid) atomic_addf(&S1[b * M + m_glob], s1c);

  // C/D layout: VGPR r, lanes 0-15 -> (M=r, N=lane); lanes 16-31 -> (M=r+8, N=lane-16)
#pragma unroll
  for (int r = 0; r < 8; ++r) {
    int mo = m_base + (khalf ? (r + 8) : r);
    if (mo < M) {
      atomic_addf(&S2[((size_t)(b * M) + mo) * F + mlane],      c0[r]);
      atomic_addf(&S2[((size_t)(b * M) + mo) * F + mlane + 16], c1[r]);
    }
  }
}

// ---------------------------------------------------------------------------
// Kernel 4: out = sigmoid(W_edge[f]*S1 + S2 + feat_dst*alpha_kk)
// ---------------------------------------------------------------------------
__global__ void __launch_bounds__(256) k_final(const float* __restrict__ er,
                                               const float* __restrict__ mx,
                                               const float* __restrict__ den,
                                               const float* __restrict__ feat_dst,
                                               const float* __restrict__ S1,
                                               const float* __restrict__ S2,
                                               const float* __restrict__ W_edge,
                                               float* __restrict__ out) {
  using namespace cfg;
  int idx = blockIdx.x * blockDim.x + threadIdx.x;
  if (idx >= B * M * F) return;
  int f  = idx & (F - 1);
  int bm = idx >> 5;  // F == 32
  float akk = __expf(leaky(2.0f * er[bm]) - mx[bm]) / den[bm];
  float val = W_edge[f] * S1[bm] + S2[idx] + feat_dst[idx] * akk;
  out[idx] = 1.0f / (1.0f + __expf(-val));
}

// ---------------------------------------------------------------------------
extern "C" void kernel_launch(void* const* d_in, const int* in_sizes, int n_in,
                              void* d_out, int out_size, void* d_ws, size_t ws_size,
                              hipStream_t stream) {
  using namespace cfg;
  (void)in_sizes; (void)n_in; (void)out_size; (void)ws_size;

  const float* raw_opes    = (const float*)d_in[0];
  const float* raw_mas     = (const float*)d_in[1];
  const float* proc_time   = (const float*)d_in[2];
  const int*   ope_ma_adj  = (const int*)  d_in[3];
  const int*   batch_idxes = (const int*)  d_in[4];
  const float* W_src       = (const float*)d_in[5];
  const float* W_dst       = (const float*)d_in[6];
  const float* W_edge      = (const float*)d_in[7];
  const float* attn_l      = (const float*)d_in[8];
  const float* attn_r      = (const float*)d_in[9];
  float* out = (float*)d_out;

  float* ws       = (float*)d_ws;
  float* feat_src = ws + off_feat_src;
  float* el       = ws + off_el;
  float* feat_dst = ws + off_feat_dst;
  float* er       = ws + off_er;
  float* mxp      = ws + off_mx;
  float* den      = ws + off_den;
  float* S1       = ws + off_S1;
  float* S2       = ws + off_S2;

  // zero the atomic accumulators (S1,S2 contiguous); capture-safe async memset
  hipMemsetAsync(S1, 0, (size_t)(B * M + B * M * F) * sizeof(float), stream);

  // 1) feature projections + attention scores
  k_feat<<<(B * O + 7) / 8, 256, 0, stream>>>(raw_opes, W_src, attn_l,
                                              feat_src, el, B * O, Ds);
  k_feat<<<(B * M + 7) / 8, 256, 0, stream>>>(raw_mas, W_dst, attn_r,
                                              feat_dst, er, B * M, Dd);

  // 2) softmax stats per (b,m)
  k_stats<<<(B * M + 7) / 8, 256, 0, stream>>>(proc_time, ope_ma_adj, batch_idxes,
                                               el, er, W_edge, attn_l, mxp, den);

  // 3) WMMA accumulation of alpha @ feat_src (+ alpha*pt)
  k_accum<<<B * MT * OCN, 32, 0, stream>>>(proc_time, ope_ma_adj, batch_idxes,
                                           el, er, mxp, den, feat_src,
                                           W_edge, attn_l, S1, S2);

  // 4) finalize with sigmoid
  k_final<<<(B * M * F + 255) / 256, 256, 0, stream>>>(er, mxp, den, feat_dst,
                                                       S1, S2, W_edge, out);
}